// FullyConnectedWithTripletLoss_43078521979195
// MI455X (gfx1250) — compile-verified
//
#include <hip/hip_runtime.h>
#include <math.h>

typedef __attribute__((ext_vector_type(2))) float v2f;
typedef __attribute__((ext_vector_type(8))) float v8f;

#define N_ROWS 4096
#define D_IN   1024
#define D_OUT  256
#define NJCH   8
#define JCH_COLS (N_ROWS / NJCH)   // 512 columns per j-chunk

// ---------------------------------------------------------------------------
// Stage 1: h = x @ W + b.  One wave (32 lanes) per 16x16 output tile.
// K-loop: 256 x V_WMMA_F32_16X16X4_F32 (fp32, exact vs reference).
// A-frag layout (ISA 7.12.2, 32-bit A 16x4): lane = M (0..15) in each half,
// VGPR0/1 = K=klo, klo+1 with klo = 0 (lanes 0-15) or 2 (lanes 16-31)
// -> contiguous float2 load per lane.
// ---------------------------------------------------------------------------
__global__ __launch_bounds__(32) void fc_wmma_kernel(const float* __restrict__ x,
                                                     const float* __restrict__ W,
                                                     const float* __restrict__ bias,
                                                     float* __restrict__ h) {
  const int lane  = threadIdx.x;
  const int row16 = lane & 15;
  const int klo   = (lane >> 4) << 1;
  const int mbase = (lane >> 4) << 3;
  const int i0 = blockIdx.x << 4;
  const int j0 = blockIdx.y << 4;

  const float* arow = x + (size_t)(i0 + row16) * D_IN + klo;
  const float* wcol = W + j0 + row16;

  v8f c = {};
#pragma unroll 4
  for (int kk = 0; kk < D_IN / 4; ++kk) {
    const int k = kk * 4;
    v2f a = *(const v2f*)(arow + k);
    v2f b;
    b.x = wcol[(size_t)(k + klo) * D_OUT];
    b.y = wcol[(size_t)(k + klo + 1) * D_OUT];
    c = __builtin_amdgcn_wmma_f32_16x16x4_f32(false, a, false, b, (short)0, c,
                                              false, false);
  }

  const float bv = bias[j0 + row16];
#pragma unroll
  for (int r = 0; r < 8; ++r) {
    // C layout: VGPR r -> M = r (lanes 0-15) / r+8 (lanes 16-31), N = lane&15
    h[(size_t)(i0 + mbase + r) * D_OUT + j0 + row16] = c[r] + bv;
  }
}

// ---------------------------------------------------------------------------
// Stage 2: sq[i] = sum_k h[i][k]^2
// ---------------------------------------------------------------------------
__global__ __launch_bounds__(256) void rowsq_kernel(const float* __restrict__ h,
                                                    float* __restrict__ sq) {
  const int row = blockIdx.x * 256 + threadIdx.x;
  const float4* hv = (const float4*)(h + (size_t)row * D_OUT);
  float s = 0.f;
#pragma unroll 8
  for (int k = 0; k < D_OUT / 4; ++k) {
    float4 v = hv[k];
    s += v.x * v.x + v.y * v.y + v.z * v.z + v.w * v.w;
  }
  sq[row] = s;
}

// ---------------------------------------------------------------------------
// Stage 3: for each 16-row anchor block and each 512-column chunk:
// Gram tile via WMMA (A-frags held in VGPRs, reused for 32 j-tiles),
// dist = sqrt(max(sq_i + sq_j - 2*G, 0)), masked running max/min,
// 16-lane xor-shuffle reduction, write per-chunk partials (no atomics).
// ---------------------------------------------------------------------------
__global__ __launch_bounds__(32) void gram_minmax_kernel(
    const float* __restrict__ h, const float* __restrict__ sq,
    const int* __restrict__ targets, float* __restrict__ pos_part,
    float* __restrict__ neg_part) {
  const int lane  = threadIdx.x;
  const int row16 = lane & 15;
  const int klo   = (lane >> 4) << 1;
  const int mbase = (lane >> 4) << 3;
  const int i0  = blockIdx.x << 4;
  const int jch = blockIdx.y;

  // Anchor fragments: 16 rows x 256 K, resident in 128 VGPRs, reused 32x.
  v2f a[D_OUT / 4];
  const float* arow = h + (size_t)(i0 + row16) * D_OUT + klo;
#pragma unroll
  for (int kk = 0; kk < D_OUT / 4; ++kk) a[kk] = *(const v2f*)(arow + kk * 4);

  float sqi[8];
  int   ti[8];
#pragma unroll
  for (int r = 0; r < 8; ++r) {
    sqi[r] = sq[i0 + mbase + r];
    ti[r]  = targets[i0 + mbase + r];
  }

  float posmax[8], negmin[8];
#pragma unroll
  for (int r = 0; r < 8; ++r) { posmax[r] = -INFINITY; negmin[r] = INFINITY; }

  const int jbeg = jch * JCH_COLS;
  for (int j0 = jbeg; j0 < jbeg + JCH_COLS; j0 += 16) {
    // B = h^T tile: element (K,N) = h[j0+N, K] -> contiguous float2 in K.
    const float* brow = h + (size_t)(j0 + row16) * D_OUT + klo;
    v8f c = {};
#pragma unroll
    for (int kk = 0; kk < D_OUT / 4; ++kk) {
      v2f b = *(const v2f*)(brow + kk * 4);
      c = __builtin_amdgcn_wmma_f32_16x16x4_f32(false, a[kk], false, b,
                                                (short)0, c, false, false);
    }

    const float sqj = sq[j0 + row16];
    const int   tj  = targets[j0 + row16];
#pragma unroll
    for (int r = 0; r < 8; ++r) {
      float d2   = sqi[r] + sqj - 2.0f * c[r];
      float dist = sqrtf(fmaxf(d2, 0.0f));
      bool  same = (ti[r] == tj);
      posmax[r] = fmaxf(posmax[r], same ? dist : -INFINITY);
      negmin[r] = fminf(negmin[r], same ? INFINITY : dist);
    }
  }

  // Reduce over N (16 lanes); xor masks 8..1 keep the two halves separate,
  // so lane 0 ends with row r and lane 16 with row r+8.
#pragma unroll
  for (int r = 0; r < 8; ++r) {
    float pm = posmax[r], nm = negmin[r];
#pragma unroll
    for (int off = 8; off >= 1; off >>= 1) {
      pm = fmaxf(pm, __shfl_xor(pm, off, 32));
      nm = fminf(nm, __shfl_xor(nm, off, 32));
    }
    if (row16 == 0) {
      const int row = i0 + mbase + r;
      pos_part[(size_t)jch * N_ROWS + row] = pm;
      neg_part[(size_t)jch * N_ROWS + row] = nm;
    }
  }
}

// ---------------------------------------------------------------------------
// Stage 4: combine chunk partials, softplus, deterministic LDS tree-sum.
// ---------------------------------------------------------------------------
__global__ __launch_bounds__(256) void loss_reduce_kernel(
    const float* __restrict__ pos_part, const float* __restrict__ neg_part,
    float* __restrict__ out) {
  __shared__ float red[256];
  const int tid = threadIdx.x;
  float acc = 0.f;
  for (int row = tid; row < N_ROWS; row += 256) {
    float pm = -INFINITY, nm = INFINITY;
#pragma unroll
    for (int c = 0; c < NJCH; ++c) {
      pm = fmaxf(pm, pos_part[(size_t)c * N_ROWS + row]);
      nm = fminf(nm, neg_part[(size_t)c * N_ROWS + row]);
    }
    const float xv = pm - nm;
    // stable softplus == log1p(exp(xv))
    acc += fmaxf(xv, 0.0f) + log1pf(expf(-fabsf(xv)));
  }
  red[tid] = acc;
  __syncthreads();
  for (int s = 128; s > 0; s >>= 1) {
    if (tid < s) red[tid] += red[tid + s];
    __syncthreads();
  }
  if (tid == 0) out[0] = red[0];
}

// ---------------------------------------------------------------------------
extern "C" void kernel_launch(void* const* d_in, const int* in_sizes, int n_in,
                              void* d_out, int out_size, void* d_ws,
                              size_t ws_size, hipStream_t stream) {
  (void)in_sizes; (void)n_in; (void)out_size; (void)ws_size;
  const float* x   = (const float*)d_in[0];
  const float* W   = (const float*)d_in[1];
  const float* b   = (const float*)d_in[2];
  const int*   tgt = (const int*)d_in[3];
  float* out = (float*)d_out;

  float* h        = (float*)d_ws;                       // 4096*256 f32 (4 MB)
  float* sq       = h + (size_t)N_ROWS * D_OUT;         // 4096 f32
  float* pos_part = sq + N_ROWS;                        // 8*4096 f32
  float* neg_part = pos_part + (size_t)NJCH * N_ROWS;   // 8*4096 f32

  fc_wmma_kernel<<<dim3(N_ROWS / 16, D_OUT / 16), 32, 0, stream>>>(x, W, b, h);
  rowsq_kernel<<<dim3(N_ROWS / 256), 256, 0, stream>>>(h, sq);
  gram_minmax_kernel<<<dim3(N_ROWS / 16, NJCH), 32, 0, stream>>>(
      h, sq, tgt, pos_part, neg_part);
  loss_reduce_kernel<<<1, 256, 0, stream>>>(pos_part, neg_part, out);
}